// GCN_35433480192644
// MI455X (gfx1250) — compile-verified
//
#include <hip/hip_runtime.h>
#include <hip/hip_bf16.h>

typedef float v2f __attribute__((ext_vector_type(2)));
typedef float v8f __attribute__((ext_vector_type(8)));

// ---------------------------------------------------------------------------
// WMMA fp32 GEMM: C[M,NC] = op(A[M,K]) @ B[K,NC] (+ bias), M multiple of 16.
// One block handles a 16-row slab; wave w computes the 16x16 tile at cols
// [16w, 16w+16). A slab staged in LDS (shared by all waves of the block).
// Uses V_WMMA_F32_16X16X4_F32 (fp32 in/out -> bit-compatible with reference).
// ---------------------------------------------------------------------------
template <int K, int NC, bool RELU_A>
__global__ __launch_bounds__(2 * NC) void gemm16(const float* __restrict__ A,
                                                 const float* __restrict__ B,
                                                 const float* __restrict__ bias,
                                                 float* __restrict__ C) {
    __shared__ float As[16 * K];
    const int tile = blockIdx.x;
    for (int idx = threadIdx.x; idx < 16 * K; idx += 2 * NC) {
        float v = A[(size_t)tile * 16 * K + idx];
        if (RELU_A) v = fmaxf(v, 0.0f);
        As[idx] = v;
    }
    __syncthreads();

    const int lane = threadIdx.x & 31;
    const int wid  = threadIdx.x >> 5;
    const int half = lane >> 4;   // 0: K pair {k0,k0+1}, 1: {k0+2,k0+3}
    const int l16  = lane & 15;
    const int colbase = wid * 16;

    v8f acc = {};
    for (int k0 = 0; k0 < K; k0 += 4) {
        const int ka = k0 + 2 * half;
        v2f a, b;
        a.x = As[l16 * K + ka];
        a.y = As[l16 * K + ka + 1];
        const float* bp = B + (size_t)ka * NC + colbase + l16;
        b.x = bp[0];
        b.y = bp[NC];
        acc = __builtin_amdgcn_wmma_f32_16x16x4_f32(false, a, false, b,
                                                    (short)0, acc, false, false);
    }

    const int col = colbase + l16;
    const float bv = bias ? bias[col] : 0.0f;
#pragma unroll
    for (int i = 0; i < 8; ++i) {
        const int row = tile * 16 + half * 8 + i;   // C/D layout: rows 0-7 / 8-15
        C[(size_t)row * NC + col] = acc[i] + bv;
    }
}

// ---------------------------------------------------------------------------
// Degree (in-degree over dst + self loop), then dinv = rsqrt(deg)
// ---------------------------------------------------------------------------
__global__ void deg_init(float* dinv, int N) {
    int i = blockIdx.x * blockDim.x + threadIdx.x;
    if (i < N) dinv[i] = 1.0f;   // self loop
}
__global__ void deg_count(const int* __restrict__ dst, float* dinv, int E) {
    int e = blockIdx.x * blockDim.x + threadIdx.x;
    if (e < E) unsafeAtomicAdd(&dinv[dst[e]], 1.0f);
}
__global__ void deg_rsqrt(float* dinv, int N) {
    int i = blockIdx.x * blockDim.x + threadIdx.x;
    if (i < N) dinv[i] = rsqrtf(dinv[i]);
}

// out[i,:] = bias + t[i,:] * dinv[i]^2   (self-loop term + bias), float4 lanes
__global__ void selfloop_bias(const float* __restrict__ t, const float* __restrict__ dinv,
                              const float* __restrict__ bias, float* __restrict__ out,
                              int N, int F4) {
    int idx = blockIdx.x * blockDim.x + threadIdx.x;   // over N*F4
    if (idx >= N * F4) return;
    int i = idx / F4;
    float di = dinv[i];
    float c = di * di;
    float4 v = ((const float4*)t)[idx];
    float4 b = ((const float4*)bias)[idx % F4];
    float4 o = {b.x + v.x * c, b.y + v.y * c, b.z + v.z * c, b.w + v.w * c};
    ((float4*)out)[idx] = o;
}

// one wave32 per edge, 128 feats = 32 lanes x float4; scatter-add stays in L2
__global__ void edge_msg128(const float* __restrict__ h, const float* __restrict__ dinv,
                            const int* __restrict__ src, const int* __restrict__ dst,
                            float* __restrict__ out, int E) {
    int e = blockIdx.x * (blockDim.x >> 5) + (threadIdx.x >> 5);
    if (e >= E) return;
    int lane = threadIdx.x & 31;
    int s = src[e], d = dst[e];
    float c = dinv[s] * dinv[d];
    float4 v = *((const float4*)(h + (size_t)s * 128) + lane);
    float* o = out + (size_t)d * 128 + lane * 4;
    unsafeAtomicAdd(o + 0, v.x * c);
    unsafeAtomicAdd(o + 1, v.y * c);
    unsafeAtomicAdd(o + 2, v.z * c);
    unsafeAtomicAdd(o + 3, v.w * c);
}

// one wave32 per edge, 64 feats = 32 lanes x float2
__global__ void edge_msg64(const float* __restrict__ h, const float* __restrict__ dinv,
                           const int* __restrict__ src, const int* __restrict__ dst,
                           float* __restrict__ out, int E) {
    int e = blockIdx.x * (blockDim.x >> 5) + (threadIdx.x >> 5);
    if (e >= E) return;
    int lane = threadIdx.x & 31;
    int s = src[e], d = dst[e];
    float c = dinv[s] * dinv[d];
    float2 v = *((const float2*)(h + (size_t)s * 64) + lane);
    float* o = out + (size_t)d * 64 + lane * 2;
    unsafeAtomicAdd(o + 0, v.x * c);
    unsafeAtomicAdd(o + 1, v.y * c);
}

// ---------------------------------------------------------------------------
// Readout: segment sum/count/max; max via monotone uint key + atomicMax
// ---------------------------------------------------------------------------
__device__ __forceinline__ unsigned fkey(float f) {
    unsigned u = __float_as_uint(f);
    return (u & 0x80000000u) ? ~u : (u | 0x80000000u);
}
__global__ void ro_init(float* sum, int* cnt, unsigned* mk, int G) {
    int i = blockIdx.x * blockDim.x + threadIdx.x;
    if (i < G * 64) { sum[i] = 0.0f; mk[i] = 0x007FFFFFu; }  // key(-inf)
    if (i < G) cnt[i] = 0;
}
__global__ void ro_accum(const float* __restrict__ h2, const int* __restrict__ batch,
                         float* sum, int* cnt, unsigned* mk, int N) {
    int idx = blockIdx.x * blockDim.x + threadIdx.x;   // over N*64
    if (idx >= N * 64) return;
    int i = idx >> 6, f = idx & 63;
    int g = batch[i];
    float v = h2[idx];
    unsafeAtomicAdd(&sum[g * 64 + f], v);
    atomicMax(&mk[g * 64 + f], fkey(v));
    if (f == 0) atomicAdd(&cnt[g], 1);
}
__global__ void ro_final(const float* sum, const int* cnt, const unsigned* mk,
                         float* r, int G) {
    int idx = blockIdx.x * blockDim.x + threadIdx.x;   // over G*64
    if (idx >= G * 64) return;
    int g = idx >> 6, f = idx & 63;
    int c = cnt[g]; if (c < 1) c = 1;
    r[(size_t)g * 128 + f] = sum[idx] / (float)c;
    unsigned k = mk[idx];
    unsigned u = (k & 0x80000000u) ? (k ^ 0x80000000u) : ~k;
    r[(size_t)g * 128 + 64 + f] = __uint_as_float(u);
}

extern "C" void kernel_launch(void* const* d_in, const int* in_sizes, int n_in,
                              void* d_out, int out_size, void* d_ws, size_t ws_size,
                              hipStream_t stream) {
    const float* x  = (const float*)d_in[0];
    const int* adj  = (const int*)d_in[1];
    const int* batch= (const int*)d_in[2];
    const float* W1 = (const float*)d_in[3];
    const float* b1 = (const float*)d_in[4];
    const float* W2 = (const float*)d_in[5];
    const float* b2 = (const float*)d_in[6];
    const float* Wm = (const float*)d_in[7];
    const float* bm = (const float*)d_in[8];
    float* out = (float*)d_out;

    const int N = in_sizes[0] / 64;   // 100000
    const int E = in_sizes[1] / 2;    // 3200000
    const int G = 1024;
    const int* src = adj;
    const int* dst = adj + E;

    float* ws   = (float*)d_ws;
    float* bufA = ws;                        // t0 [N,128]; later t1 [N,64]
    float* bufB = ws + (size_t)N * 128;      // h  [N,128]; later h2 [N,64]
    float* dinv = bufB + (size_t)N * 128;    // [N]
    float* sum  = dinv + N;                  // [G,64]
    int*   cnt  = (int*)(sum + (size_t)G * 64);       // [G]
    unsigned* mk = (unsigned*)(cnt + G);              // [G,64]
    float* r    = (float*)(mk + (size_t)G * 64);      // [G,128]

    // ---- layer 1: t0 = x @ W1 (WMMA f32) ----
    gemm16<64, 128, false><<<N / 16, 256, 0, stream>>>(x, W1, nullptr, bufA);

    // ---- symmetric norm ----
    deg_init <<<(N + 255) / 256, 256, 0, stream>>>(dinv, N);
    deg_count<<<(E + 255) / 256, 256, 0, stream>>>(dst, dinv, E);
    deg_rsqrt<<<(N + 255) / 256, 256, 0, stream>>>(dinv, N);

    // ---- h = b1 + selfloop + scatter-add over edges (L2-resident) ----
    selfloop_bias<<<(N * 32 + 255) / 256, 256, 0, stream>>>(bufA, dinv, b1, bufB, N, 32);
    edge_msg128 <<<(E + 7) / 8, 256, 0, stream>>>(bufA, dinv, src, dst, bufB, E);

    // ---- layer 2: t1 = relu(h) @ W2 (relu folded into A-load) ----
    gemm16<128, 64, true><<<N / 16, 128, 0, stream>>>(bufB, W2, nullptr, bufA);

    // ---- h2 = b2 + selfloop + scatter-add over edges ----
    selfloop_bias<<<(N * 16 + 255) / 256, 256, 0, stream>>>(bufA, dinv, b2, bufB, N, 16);
    edge_msg64  <<<(E + 7) / 8, 256, 0, stream>>>(bufA, dinv, src, dst, bufB, E);

    // ---- readout mean|max -> r [G,128] ----
    ro_init <<<(G * 64 + 255) / 256, 256, 0, stream>>>(sum, cnt, mk, G);
    ro_accum<<<(N * 64 + 255) / 256, 256, 0, stream>>>(bufB, batch, sum, cnt, mk, N);
    ro_final<<<(G * 64 + 255) / 256, 256, 0, stream>>>(sum, cnt, mk, r, G);

    // ---- final: out = r @ Wm + bm (WMMA f32) ----
    gemm16<128, 64, false><<<G / 16, 128, 0, stream>>>(r, Wm, bm, out);
}